// TransitionDown_40303973105895
// MI455X (gfx1250) — compile-verified
//
#include <hip/hip_runtime.h>
#include <hip/hip_bf16.h>

typedef __attribute__((ext_vector_type(2))) float v2f;
typedef __attribute__((ext_vector_type(8))) float v8f;

#define Bc 8
#define Nc 8192
#define INC_F 64
#define OUTC 128
#define KNEAR 16
#define Mc 2048
#define INC 68        // 4 + 64 concat channels

// ---------------------------------------------------------------------------
// Kernel 1: farthest point sampling. One block per batch, 1024 threads,
// 8 points/thread in registers. Argmax reduction also carries the winning
// point's coordinates so the serial loop needs no global memory at all.
// ---------------------------------------------------------------------------
#define FPS_T 1024
#define PPT (Nc / FPS_T)   // 8

__global__ __launch_bounds__(FPS_T) void fps_kernel(const float* __restrict__ xyzp,
                                                    int* __restrict__ fps_idx) {
  __shared__ float rv[32], rx[32], ry[32], rz[32];
  __shared__ int ri[32];
  __shared__ float bxs, bys, bzs;
  __shared__ int bis;

  const int b = blockIdx.x;
  const int t = threadIdx.x;
  const int lane = t & 31;
  const int wave = t >> 5;
  const float* base = xyzp + (size_t)b * Nc * 4;

  float px[PPT], py[PPT], pz[PPT], dist[PPT];
#pragma unroll
  for (int i = 0; i < PPT; ++i) {
    const int p = t + i * FPS_T;
    const float4 v = *(const float4*)(base + (size_t)p * 4);
    px[i] = v.x; py[i] = v.y; pz[i] = v.z;
    dist[i] = __builtin_inff();
  }

  int farthest = 0;
  float cx = base[0], cy = base[1], cz = base[2];  // point 0 (broadcast load)

  for (int k = 0; k < Mc; ++k) {
    if (t == 0) fps_idx[b * Mc + k] = farthest;

    float bv = -1.0f, bpx = 0.f, bpy = 0.f, bpz = 0.f;
    int bidx = 0;
#pragma unroll
    for (int i = 0; i < PPT; ++i) {
      const float dx = px[i] - cx, dy = py[i] - cy, dz = pz[i] - cz;
      const float d = dx * dx + dy * dy + dz * dz;
      dist[i] = fminf(dist[i], d);
      if (dist[i] > bv) { bv = dist[i]; bidx = t + i * FPS_T; bpx = px[i]; bpy = py[i]; bpz = pz[i]; }
    }
    // wave32 argmax (value desc, index asc tie-break), carrying coordinates
#pragma unroll
    for (int off = 16; off > 0; off >>= 1) {
      const float ov = __shfl_xor(bv, off, 32);
      const int   oi = __shfl_xor(bidx, off, 32);
      const float ox = __shfl_xor(bpx, off, 32);
      const float oy = __shfl_xor(bpy, off, 32);
      const float oz = __shfl_xor(bpz, off, 32);
      if (ov > bv || (ov == bv && oi < bidx)) { bv = ov; bidx = oi; bpx = ox; bpy = oy; bpz = oz; }
    }
    if (lane == 0) { rv[wave] = bv; ri[wave] = bidx; rx[wave] = bpx; ry[wave] = bpy; rz[wave] = bpz; }
    __syncthreads();
    if (wave == 0) {
      bv = rv[lane]; bidx = ri[lane]; bpx = rx[lane]; bpy = ry[lane]; bpz = rz[lane];
#pragma unroll
      for (int off = 16; off > 0; off >>= 1) {
        const float ov = __shfl_xor(bv, off, 32);
        const int   oi = __shfl_xor(bidx, off, 32);
        const float ox = __shfl_xor(bpx, off, 32);
        const float oy = __shfl_xor(bpy, off, 32);
        const float oz = __shfl_xor(bpz, off, 32);
        if (ov > bv || (ov == bv && oi < bidx)) { bv = ov; bidx = oi; bpx = ox; bpy = oy; bpz = oz; }
      }
      if (lane == 0) { bis = bidx; bxs = bpx; bys = bpy; bzs = bpz; }
    }
    __syncthreads();
    farthest = bis; cx = bxs; cy = bys; cz = bzs;
  }
}

// ---------------------------------------------------------------------------
// Kernel 2: kNN (top-16 smallest d2, lowest-index tie-break) + new_xyzp out.
// One thread per query; points tiled through LDS with precomputed |p|^2.
// ---------------------------------------------------------------------------
#define KNN_T 256
#define TILE 2048

__global__ __launch_bounds__(KNN_T) void knn_kernel(const float* __restrict__ xyzp,
                                                    const int* __restrict__ fps_idx,
                                                    int* __restrict__ knn_idx,
                                                    float* __restrict__ out_xyzp) {
  __shared__ float sx[TILE], sy[TILE], sz[TILE], sp[TILE];
  const int b = blockIdx.x / (Mc / KNN_T);
  const int m = (blockIdx.x % (Mc / KNN_T)) * KNN_T + threadIdx.x;
  const float* base = xyzp + (size_t)b * Nc * 4;

  const int qi = fps_idx[b * Mc + m];
  const float4 q4 = *(const float4*)(base + (size_t)qi * 4);
  const float qx = q4.x, qy = q4.y, qz = q4.z;
  const float qq = qx * qx + qy * qy + qz * qz;

  float bd[KNEAR];
  int bi[KNEAR];
#pragma unroll
  for (int j = 0; j < KNEAR; ++j) { bd[j] = __builtin_inff(); bi[j] = 0; }

  for (int t0 = 0; t0 < Nc; t0 += TILE) {
    __syncthreads();
    for (int i = threadIdx.x; i < TILE; i += KNN_T) {
      const float4 p = *(const float4*)(base + (size_t)(t0 + i) * 4);
      sx[i] = p.x; sy[i] = p.y; sz[i] = p.z;
      sp[i] = p.x * p.x + p.y * p.y + p.z * p.z;
    }
    __syncthreads();
    for (int i = 0; i < TILE; ++i) {
      const float d = qq - 2.0f * (qx * sx[i] + qy * sy[i] + qz * sz[i]) + sp[i];
      if (d < bd[KNEAR - 1]) {
        const int pid = t0 + i;
        bool placed = false;
#pragma unroll
        for (int s = KNEAR - 1; s >= 1; --s) {
          if (!placed) {
            if (d < bd[s - 1]) { bd[s] = bd[s - 1]; bi[s] = bi[s - 1]; }
            else               { bd[s] = d;         bi[s] = pid; placed = true; }
          }
        }
        if (!placed) { bd[0] = d; bi[0] = pid; }
      }
    }
  }

  int* ko = knn_idx + (size_t)(b * Mc + m) * KNEAR;
#pragma unroll
  for (int j = 0; j < KNEAR; ++j) ko[j] = bi[j];

  const float4 nx = *(const float4*)(base + (size_t)bi[0] * 4);
  float* o = out_xyzp + (size_t)(b * Mc + m) * 4;
  o[0] = nx.x; o[1] = nx.y; o[2] = nx.z; o[3] = nx.w;
}

// ---------------------------------------------------------------------------
// Kernel 3: gather + [16x68]@[68x128] per group via V_WMMA_F32_16X16X4_F32,
// fused BN + ReLU + max-over-16-neighbors.
// One wave per group; 4 waves/block; W staged once per block in LDS.
// ---------------------------------------------------------------------------
#define WSTR 136            // W LDS row stride (floats): 2*136 % 64 = 16 -> no bank conflict
#define XSTR 76             // X LDS row stride (floats): 12r mod 64 distinct, 16B aligned
#define K3_WAVES 4
#define GRP_PER_WAVE 16
#define GRP_PER_BLOCK (K3_WAVES * GRP_PER_WAVE)

__global__ __launch_bounds__(128) void group_gemm_kernel(
    const float* __restrict__ xyzp, const float* __restrict__ feat,
    const int* __restrict__ knn_idx, const float* __restrict__ Wm,
    const float* __restrict__ bias, const float* __restrict__ gamma,
    const float* __restrict__ beta, const float* __restrict__ rmean,
    const float* __restrict__ rvar, float* __restrict__ out_feat) {
  __shared__ float Ws[INC * WSTR];                 // ~36.1 KB
  __shared__ float Xs[K3_WAVES][16 * XSTR];        // ~19.0 KB
  __shared__ float sS[OUTC], sT[OUTC];

  const int t = threadIdx.x;
  const int lane = t & 31;
  const int wave = t >> 5;
  const int r16 = lane & 15;
  const int half = lane >> 4;

  // Stage W[68,128] -> LDS (stride WSTR)
  for (int e = t; e < INC * (OUTC / 4); e += 128) {
    const int r = e >> 5, c4 = e & 31;
    const float4 wv = *(const float4*)(Wm + (size_t)r * OUTC + c4 * 4);
    float* dst = &Ws[r * WSTR + c4 * 4];
    dst[0] = wv.x; dst[1] = wv.y; dst[2] = wv.z; dst[3] = wv.w;
  }
  if (t < OUTC) {
    const float s = gamma[t] * rsqrtf(rvar[t] + 1e-5f);
    sS[t] = s;
    sT[t] = (bias[t] - rmean[t]) * s + beta[t];
  }
  __syncthreads();

  for (int it = 0; it < GRP_PER_WAVE; ++it) {
    const int g = blockIdx.x * GRP_PER_BLOCK + wave * GRP_PER_WAVE + it;
    const int b = g / Mc;
    const int* kn = knn_idx + (size_t)g * KNEAR;
    const float* xb = xyzp + (size_t)b * Nc * 4;
    const float* fb = feat + (size_t)b * Nc * INC_F;
    float* xw = &Xs[wave][0];

    const int n0 = kn[0];
    const float4 c0 = *(const float4*)(xb + (size_t)n0 * 4);

    // Gather 16 rows x 17 float4 each: [xyzp_norm(4) | features(64)]
    for (int e = lane; e < 16 * 17; e += 32) {
      const int r = e / 17, q = e % 17;
      const int ni = kn[r];
      float4 v;
      if (q == 0) {
        const float4 pv = *(const float4*)(xb + (size_t)ni * 4);
        v.x = pv.x - c0.x; v.y = pv.y - c0.y; v.z = pv.z - c0.z; v.w = pv.w - c0.w;
      } else {
        v = *(const float4*)(fb + (size_t)ni * INC_F + (q - 1) * 4);
      }
      *(float4*)(xw + r * XSTR + q * 4) = v;
    }
    __syncthreads();

    // Preload all 17 A fragments (K=4 each): lane holds row r16, K = 4*ks + 2*half + {0,1}
    v2f a[17];
#pragma unroll
    for (int ks = 0; ks < 17; ++ks) {
      const int kk = ks * 4 + half * 2;
      a[ks].x = xw[r16 * XSTR + kk];
      a[ks].y = xw[r16 * XSTR + kk + 1];
    }

    float rowmax[8];
#pragma unroll
    for (int n = 0; n < 8; ++n) {
      v8f acc = {0.f, 0.f, 0.f, 0.f, 0.f, 0.f, 0.f, 0.f};
#pragma unroll
      for (int ks = 0; ks < 17; ++ks) {
        const int kk = ks * 4 + half * 2;
        v2f bb;
        bb.x = Ws[kk * WSTR + n * 16 + r16];
        bb.y = Ws[(kk + 1) * WSTR + n * 16 + r16];
        acc = __builtin_amdgcn_wmma_f32_16x16x4_f32(
            false, a[ks], false, bb, (short)0, acc, false, false);
      }
      // Fused BN + ReLU + max over the 16 neighbor rows.
      const int col = n * 16 + r16;
      const float s = sS[col], tt = sT[col];
      float mx = acc[0] * s + tt;
#pragma unroll
      for (int v = 1; v < 8; ++v) mx = fmaxf(mx, acc[v] * s + tt);
      const float om = __shfl_xor(mx, 16, 32);     // combine row halves
      rowmax[n] = fmaxf(fmaxf(mx, om), 0.0f);      // relu(max) == max(relu)
    }

    if (half == 0) {
      float* o = out_feat + (size_t)g * OUTC;
#pragma unroll
      for (int n = 0; n < 8; ++n) o[n * 16 + r16] = rowmax[n];
    }
    __syncthreads();
  }
}

// ---------------------------------------------------------------------------
extern "C" void kernel_launch(void* const* d_in, const int* in_sizes, int n_in,
                              void* d_out, int out_size, void* d_ws, size_t ws_size,
                              hipStream_t stream) {
  const float* xyzp  = (const float*)d_in[0];
  const float* feat  = (const float*)d_in[1];
  const float* Wm    = (const float*)d_in[2];
  const float* bias  = (const float*)d_in[3];
  const float* gamma = (const float*)d_in[4];
  const float* beta  = (const float*)d_in[5];
  const float* rmean = (const float*)d_in[6];
  const float* rvar  = (const float*)d_in[7];

  float* out_xyzp = (float*)d_out;                       // [B, M, 4]
  float* out_feat = (float*)d_out + (size_t)Bc * Mc * 4; // [B, M, 128]

  int* fps = (int*)d_ws;                                            // B*M ints
  int* knn = (int*)((char*)d_ws + (size_t)Bc * Mc * sizeof(int));   // B*M*16 ints

  fps_kernel<<<Bc, FPS_T, 0, stream>>>(xyzp, fps);
  knn_kernel<<<Bc * (Mc / KNN_T), KNN_T, 0, stream>>>(xyzp, fps, knn, out_xyzp);
  group_gemm_kernel<<<(Bc * Mc) / GRP_PER_BLOCK, 128, 0, stream>>>(
      xyzp, feat, knn, Wm, bias, gamma, beta, rmean, rvar, out_feat);
}